// GCNIILayer_34179349742037
// MI455X (gfx1250) — compile-verified
//
#include <hip/hip_runtime.h>
#include <hip/hip_bf16.h>
#include <math.h>

typedef __attribute__((ext_vector_type(2))) float v2f;
typedef __attribute__((ext_vector_type(8))) float v8f;

#define D_FEAT 64

// ---------------------------------------------------------------------------
// Kernel 1: zero the scratch (agg rows + degree array)
// ---------------------------------------------------------------------------
__global__ void k_zero(float* p, long n) {
    long i = (long)blockIdx.x * blockDim.x + threadIdx.x;
    if (i < n) p[i] = 0.0f;
}

// ---------------------------------------------------------------------------
// Kernel 2: in-degree histogram (float, via global_atomic_add_f32)
// ---------------------------------------------------------------------------
__global__ void k_degree(const int* __restrict__ dst, float* __restrict__ degs, int n_edges) {
    int i = blockIdx.x * blockDim.x + threadIdx.x;
    if (i < n_edges) atomicAdd(&degs[dst[i]], 1.0f);
}

// ---------------------------------------------------------------------------
// Kernel 3: norm = max(deg,1)^-0.5
// ---------------------------------------------------------------------------
__global__ void k_norm(const float* __restrict__ degs, float* __restrict__ norm, int n_nodes) {
    int i = blockIdx.x * blockDim.x + threadIdx.x;
    if (i < n_nodes) norm[i] = rsqrtf(fmaxf(degs[i], 1.0f));
}

// ---------------------------------------------------------------------------
// Kernel 4: edge scatter — one wave32 per edge.
// agg[dst] += feat[src] * norm[src]; lane covers features lane and lane+32.
// Coalesced 4B loads + coalesced f32 atomics (two per lane).
// ---------------------------------------------------------------------------
__global__ void k_edge_scatter(const float* __restrict__ feat,
                               const float* __restrict__ norm,
                               const int* __restrict__ src,
                               const int* __restrict__ dst,
                               float* __restrict__ agg,
                               int n_edges) {
    int wid  = (blockIdx.x * blockDim.x + threadIdx.x) >> 5;
    int lane = threadIdx.x & 31;
    if (wid >= n_edges) return;
    int s = src[wid];
    int d = dst[wid];
    float ns = norm[s];
    const float* frow = feat + (long)s * D_FEAT;
    float v0 = frow[lane]      * ns;
    float v1 = frow[lane + 32] * ns;
    float* arow = agg + (long)d * D_FEAT;
    atomicAdd(&arow[lane],      v0);
    atomicAdd(&arow[lane + 32], v1);
}

// ---------------------------------------------------------------------------
// Kernel 5: fused finalize.
//   fs      = agg*norm*(1-alpha) + feat_0*alpha        (16x64 tile in LDS)
//   out     = (1-beta)*fs + beta*(fs @ W)              (W is 64x64)
// Block = 128 threads = 4 waves; wave w owns output columns [16w, 16w+16).
// GEMM: 16 chained V_WMMA_F32_16X16X4_F32 over K=64.
// ---------------------------------------------------------------------------
__global__ void __launch_bounds__(128)
k_fused_wmma(const float* __restrict__ agg,
             const float* __restrict__ norm,
             const float* __restrict__ feat0,
             const float* __restrict__ weight,
             const float* __restrict__ alphap,
             const float* __restrict__ lamdap,
             const int*  __restrict__ layerp,
             float* __restrict__ out,
             int n_nodes) {
    __shared__ float fsum[16][68];   // stride 68 -> conflict-free row access

    const int lane = threadIdx.x & 31;
    const int wave = threadIdx.x >> 5;      // 0..3
    const int row0 = blockIdx.x * 16;

    const float alpha = *alphap;
    const float beta  = logf(*lamdap / (float)(*layerp) + 1.0f);

    // Stage feat_sum tile into LDS (128 threads x 8 elements)
    for (int idx = threadIdx.x; idx < 16 * D_FEAT; idx += 128) {
        int r = idx >> 6;
        int c = idx & 63;
        int node = row0 + r;
        float v = 0.0f;
        if (node < n_nodes) {
            v = agg[(long)node * D_FEAT + c] * norm[node] * (1.0f - alpha)
              + feat0[(long)node * D_FEAT + c] * alpha;
        }
        fsum[r][c] = v;
    }
    __syncthreads();

    const int half  = lane >> 4;      // 0: K even pair base 0 / rows 0-7 of C
    const int l16   = lane & 15;
    const int kbase = half * 2;       // A/B K base for this half-wave
    const int n0    = wave * 16;      // output column tile

    v8f c = {};
    #pragma unroll
    for (int k = 0; k < 16; ++k) {
        int kk = k * 4;
        v2f a, b;
        // A 16x4 layout: lane holds row M=l16, K = kbase, kbase+1
        a.x = fsum[l16][kk + kbase];
        a.y = fsum[l16][kk + kbase + 1];
        // B 4x16 layout: lane holds col N=l16, K = kbase, kbase+1
        b.x = weight[(kk + kbase)     * D_FEAT + n0 + l16];
        b.y = weight[(kk + kbase + 1) * D_FEAT + n0 + l16];
        c = __builtin_amdgcn_wmma_f32_16x16x4_f32(
                /*neg_a=*/false, a, /*neg_b=*/false, b,
                /*c_mod=*/(short)0, c, /*reuse_a=*/false, /*reuse_b=*/false);
    }

    // C/D layout: VGPR r -> row (r + 8*half), col = n0 + l16
    #pragma unroll
    for (int r = 0; r < 8; ++r) {
        int m    = r + half * 8;
        int node = row0 + m;
        if (node < n_nodes) {
            int col  = n0 + l16;
            float fs = fsum[m][col];
            out[(long)node * D_FEAT + col] = (1.0f - beta) * fs + beta * c[r];
        }
    }
}

// ---------------------------------------------------------------------------
extern "C" void kernel_launch(void* const* d_in, const int* in_sizes, int n_in,
                              void* d_out, int out_size, void* d_ws, size_t ws_size,
                              hipStream_t stream) {
    const float* feat   = (const float*)d_in[0];
    const float* feat0  = (const float*)d_in[1];
    const float* weight = (const float*)d_in[2];
    const int*   src    = (const int*)  d_in[3];
    const int*   dst    = (const int*)  d_in[4];
    const float* alphap = (const float*)d_in[5];
    const float* lamdap = (const float*)d_in[6];
    const int*   layerp = (const int*)  d_in[7];
    float*       out    = (float*)d_out;

    const int n_nodes = in_sizes[0] / D_FEAT;
    const int n_edges = in_sizes[3];

    // Workspace layout
    float* agg  = (float*)d_ws;                      // n_nodes * 64
    float* degs = agg  + (long)n_nodes * D_FEAT;     // n_nodes
    float* nrm  = degs + n_nodes;                    // n_nodes

    // 1) zero agg + degs
    long nz = (long)n_nodes * D_FEAT + n_nodes;
    k_zero<<<(unsigned)((nz + 255) / 256), 256, 0, stream>>>((float*)d_ws, nz);

    // 2) degree histogram
    k_degree<<<(n_edges + 255) / 256, 256, 0, stream>>>(dst, degs, n_edges);

    // 3) norm
    k_norm<<<(n_nodes + 255) / 256, 256, 0, stream>>>(degs, nrm, n_nodes);

    // 4) edge scatter (one wave32 per edge; 8 edges per 256-thread block)
    k_edge_scatter<<<(n_edges + 7) / 8, 256, 0, stream>>>(feat, nrm, src, dst, agg, n_edges);

    // 5) fused finalize + WMMA GEMM (16 nodes per block)
    k_fused_wmma<<<(n_nodes + 15) / 16, 128, 0, stream>>>(
        agg, nrm, feat0, weight, alphap, lamdap, layerp, out, n_nodes);
}